// CalciumAttentionModule_76398878261811
// MI455X (gfx1250) — compile-verified
//
#include <hip/hip_runtime.h>

typedef __attribute__((ext_vector_type(2))) float v2f;
typedef __attribute__((ext_vector_type(8))) float v8f;

#define BATCH  256
#define NFEAT  512
#define DMODEL 1024
#define DHALF  512
#define LN_EPS 1e-5f

// ---------------------------------------------------------------------------
// f32 WMMA GEMM: D = A(MxK) x B(KxN) (+bias) with fused epilogues.
// One wave computes a 16 x (16*NB) output stripe via V_WMMA_F32_16X16X4_F32.
// Software-pipelined with a PF-deep register ring: PF stages (4 K each) of
// A/B fragments are kept in flight so WMMA issue overlaps global-load latency
// (required: ~1 wave/SIMD at this problem size, so ILP is the only latency
// hiding available).
//   BT=true : B stored [N,K] row-major (weight matrix, use B^T)
//   BT=false: B stored [K,N] row-major
//   MODE 0: d += bias; store phi; store (cos d, sin d) interleaved
//   MODE 1: d += bias; d = sigmoid(d); store
//   MODE 2: d = d * gate[m, n%512] + phi[m,n]; store
// ---------------------------------------------------------------------------
template<int KDIM, bool BT, int MODE, int NB, int PF>
__global__ __launch_bounds__(32)
void wmma_gemm_f32(const float* __restrict__ A, const float* __restrict__ B,
                   const float* __restrict__ bias, const float* gate,
                   const float* phi,
                   float* __restrict__ out0, float* __restrict__ out1, int N)
{
    constexpr int STAGES = KDIM / 4;
    static_assert(STAGES % PF == 0, "K stages must divide pipeline depth");

    const int tilesN = N / (16 * NB);
    const int tile   = blockIdx.x;
    const int tm     = (tile / tilesN) << 4;            // output row base
    const int tn     = (tile % tilesN) * (16 * NB);     // output col base
    const int lane   = threadIdx.x;                     // 0..31, wave32
    const int half   = lane >> 4;                       // 0 or 1
    const int lm     = lane & 15;

    v8f acc[NB];
#pragma unroll
    for (int nt = 0; nt < NB; ++nt) {
        v8f z = {0.f, 0.f, 0.f, 0.f, 0.f, 0.f, 0.f, 0.f};
        acc[nt] = z;
    }

    // Per-lane base pointers with the half-wave K offset (2*half) folded in.
    const float* arow = A + (size_t)(tm + lm) * KDIM + 2 * half;
    const float* brow[NB];
#pragma unroll
    for (int nt = 0; nt < NB; ++nt) {
        const int n = tn + nt * 16 + lm;
        brow[nt] = BT ? (B + (size_t)n * KDIM + 2 * half)
                      : (B + (size_t)(2 * half) * N + n);
    }

    // PF-deep fragment ring (all indices compile-time under full unroll).
    v2f abuf[PF];
    v2f bbuf[PF][NB];

    auto load_stage = [&](int k, int s) {
        abuf[s].x = arow[k + 0];
        abuf[s].y = arow[k + 1];
#pragma unroll
        for (int nt = 0; nt < NB; ++nt) {
            if (BT) {
                bbuf[s][nt].x = brow[nt][k + 0];
                bbuf[s][nt].y = brow[nt][k + 1];
            } else {
                const float* p = brow[nt] + (size_t)k * N;
                bbuf[s][nt].x = p[0];
                bbuf[s][nt].y = p[N];
            }
        }
    };

    // Prologue: fill the pipeline (PF stages = 3*PF b64 loads in flight).
#pragma unroll
    for (int s = 0; s < PF; ++s) load_stage(4 * s, s);

    // Steady state: consume stage s (NB WMMAs sharing the A fragment),
    // immediately refill it PF stages ahead.
    for (int it = 0; it < STAGES / PF - 1; ++it) {
        const int kbase = it * (4 * PF);
#pragma unroll
        for (int s = 0; s < PF; ++s) {
#pragma unroll
            for (int nt = 0; nt < NB; ++nt)
                acc[nt] = __builtin_amdgcn_wmma_f32_16x16x4_f32(
                              false, abuf[s], false, bbuf[s][nt], (short)0,
                              acc[nt], false, false);
            load_stage(kbase + 4 * (PF + s), s);
        }
    }

    // Drain.
#pragma unroll
    for (int s = 0; s < PF; ++s) {
#pragma unroll
        for (int nt = 0; nt < NB; ++nt)
            acc[nt] = __builtin_amdgcn_wmma_f32_16x16x4_f32(
                          false, abuf[s], false, bbuf[s][nt], (short)0,
                          acc[nt], false, false);
    }

    // Epilogue: element v of tile nt lives at (m = tm + v + 8*half, n below)
#pragma unroll
    for (int nt = 0; nt < NB; ++nt) {
        const int n = tn + nt * 16 + lm;
#pragma unroll
        for (int v = 0; v < 8; ++v) {
            const int    m   = tm + v + 8 * half;
            const size_t idx = (size_t)m * N + n;
            float d = acc[nt][v];
            if (MODE == 0) {
                d += bias[n];
                out0[idx] = d;                      // phi
                out1[2 * idx + 0] = __cosf(d);      // unit phase vector, interleaved
                out1[2 * idx + 1] = __sinf(d);
            } else if (MODE == 1) {
                d += bias[n];
                out0[idx] = 1.0f / (1.0f + __expf(-d));   // sigmoid gate
            } else {
                const float g = gate[(size_t)m * DHALF + (n & (DHALF - 1))];
                out0[idx] = d * g + phi[idx];       // features_pre = coupled*gate + phi
            }
        }
    }
}

// ---------------------------------------------------------------------------
// ca_update[b,i] = mean_j |c_i c_j + s_i s_j| ;  ca = 0.95*prev + 0.05*update
// One block per batch row; (c,s) row staged in LDS as float4 (two pairs per
// ds_load_b128, same-address broadcast across lanes); 4 i-values per thread.
// Inner loop is pure full-rate VALU: v_mul + v_fma + v_add with |x| modifier.
// ---------------------------------------------------------------------------
__global__ __launch_bounds__(256)
void ca_coherence_kernel(const float4* __restrict__ cs,
                         const float* __restrict__ prev_ca,
                         float* __restrict__ ca_out, float* __restrict__ ca_ws)
{
    __shared__ float4 u4[DMODEL / 2];
    const int b = blockIdx.x;
    const int t = threadIdx.x;

    const float4* row = cs + (size_t)b * (DMODEL / 2);
    for (int j = t; j < DMODEL / 2; j += 256) u4[j] = row[j];
    __syncthreads();

    const float2* u2 = (const float2*)u4;
    float ci[4], si[4], acc[4];
#pragma unroll
    for (int q = 0; q < 4; ++q) {
        float2 w = u2[t + 256 * q];
        ci[q] = w.x; si[q] = w.y; acc[q] = 0.f;
    }

#pragma unroll 4
    for (int j = 0; j < DMODEL / 2; ++j) {
        const float4 w = u4[j];          // (c_{2j}, s_{2j}, c_{2j+1}, s_{2j+1})
#pragma unroll
        for (int q = 0; q < 4; ++q) {
            acc[q] += fabsf(fmaf(ci[q], w.x, si[q] * w.y));
            acc[q] += fabsf(fmaf(ci[q], w.z, si[q] * w.w));
        }
    }

#pragma unroll
    for (int q = 0; q < 4; ++q) {
        const size_t idx = (size_t)b * DMODEL + t + 256 * q;
        const float ca = prev_ca[idx] * 0.95f + acc[q] * (0.05f / DMODEL);
        ca_out[idx] = ca;   // tuple output #2
        ca_ws[idx]  = ca;   // feeds gate GEMM
    }
}

// ---------------------------------------------------------------------------
// Row-wise LayerNorm over d=1024: 256 threads/block, 4 elems/thread.
// ---------------------------------------------------------------------------
__global__ __launch_bounds__(256)
void layernorm_kernel(const float* __restrict__ fpre,
                      const float* __restrict__ gamma,
                      const float* __restrict__ beta,
                      float* __restrict__ out)
{
    __shared__ float rs[8], rq[8];
    const int b = blockIdx.x;
    const int t = threadIdx.x;
    const float* row = fpre + (size_t)b * DMODEL;

    float x[4], sum = 0.f, sumsq = 0.f;
#pragma unroll
    for (int q = 0; q < 4; ++q) {
        x[q] = row[t + 256 * q];
        sum += x[q];
        sumsq += x[q] * x[q];
    }
    // wave32 reduction
#pragma unroll
    for (int off = 16; off > 0; off >>= 1) {
        sum   += __shfl_xor(sum,   off, 32);
        sumsq += __shfl_xor(sumsq, off, 32);
    }
    const int wave = t >> 5;
    if ((t & 31) == 0) { rs[wave] = sum; rq[wave] = sumsq; }
    __syncthreads();
    sum = 0.f; sumsq = 0.f;
#pragma unroll
    for (int w = 0; w < 8; ++w) { sum += rs[w]; sumsq += rq[w]; }

    const float mu   = sum * (1.0f / DMODEL);
    const float var  = sumsq * (1.0f / DMODEL) - mu * mu;
    const float rstd = rsqrtf(var + LN_EPS);

#pragma unroll
    for (int q = 0; q < 4; ++q) {
        const int i = t + 256 * q;
        out[(size_t)b * DMODEL + i] = (x[q] - mu) * rstd * gamma[i] + beta[i];
    }
}

// ---------------------------------------------------------------------------
extern "C" void kernel_launch(void* const* d_in, const int* in_sizes, int n_in,
                              void* d_out, int out_size, void* d_ws, size_t ws_size,
                              hipStream_t stream)
{
    const float* x       = (const float*)d_in[0];   // [256, 512]
    const float* prev_ca = (const float*)d_in[1];   // [256, 1024]
    const float* Wp      = (const float*)d_in[2];   // [1024, 512]
    const float* bp      = (const float*)d_in[3];   // [1024]
    const float* Wg      = (const float*)d_in[4];   // [512, 1024]
    const float* bg      = (const float*)d_in[5];   // [512]
    const float* W       = (const float*)d_in[6];   // [1024, 1024]
    const float* gamma   = (const float*)d_in[7];   // [1024]
    const float* beta    = (const float*)d_in[8];   // [1024]

    float* out = (float*)d_out;
    float* out_features = out;                              // [256,1024]
    float* out_ca       = out + (size_t)BATCH * DMODEL;     // [256,1024]
    float* out_W        = out + (size_t)2 * BATCH * DMODEL; // [1024,1024]

    // workspace layout (floats)
    float* ws   = (float*)d_ws;
    float* phi  = ws;                                       // 256*1024
    float* cs   = phi  + (size_t)BATCH * DMODEL;            // 256*1024*2 (float2 pairs)
    float* ca   = cs   + (size_t)BATCH * DMODEL * 2;        // 256*1024
    float* gate = ca   + (size_t)BATCH * DMODEL;            // 256*512
    float* fpre = gate + (size_t)BATCH * DHALF;             // 256*1024

    constexpr int NB = 2;   // column tiles per wave (A-fragment reuse)
    constexpr int PF = 8;   // software-pipeline depth (k-stages in flight)

    // 1) phi = x @ Wp^T + bp ; also stash (cos,sin) of phi
    {
        const int tiles = (BATCH / 16) * (DMODEL / (16 * NB));   // 512
        wmma_gemm_f32<NFEAT, true, 0, NB, PF><<<tiles, 32, 0, stream>>>(
            x, Wp, bp, nullptr, nullptr, phi, cs, DMODEL);
    }

    // 2) pairwise |cos| coherence -> ca
    ca_coherence_kernel<<<BATCH, 256, 0, stream>>>(
        (const float4*)cs, prev_ca, out_ca, ca);

    // 3) gate = sigmoid(ca @ Wg^T + bg)
    {
        const int tiles = (BATCH / 16) * (DHALF / (16 * NB));    // 256
        wmma_gemm_f32<DMODEL, true, 1, NB, PF><<<tiles, 32, 0, stream>>>(
            ca, Wg, bg, nullptr, nullptr, gate, nullptr, DHALF);
    }

    // 4) fpre = (phi @ W) * gate_tiled + phi
    {
        const int tiles = (BATCH / 16) * (DMODEL / (16 * NB));   // 512
        wmma_gemm_f32<DMODEL, false, 2, NB, PF><<<tiles, 32, 0, stream>>>(
            phi, W, nullptr, gate, phi, fpre, nullptr, DMODEL);
    }

    // 5) features = LayerNorm(fpre)
    layernorm_kernel<<<BATCH, 256, 0, stream>>>(fpre, gamma, beta, out_features);

    // 6) third tuple element: W passthrough
    hipMemcpyAsync(out_W, W, (size_t)DMODEL * DMODEL * sizeof(float),
                   hipMemcpyDeviceToDevice, stream);
}